// LightingProbes_57440892617286
// MI455X (gfx1250) — compile-verified
//
#include <hip/hip_runtime.h>
#include <math.h>

typedef __attribute__((ext_vector_type(2))) float v2f;
typedef __attribute__((ext_vector_type(8))) float v8f;

#define NPROBE   125
#define PTILES   8            // 125 probes padded to 8 tiles of 16
#define SH_F     48           // 16 coeffs * 3 channels
#define BIGD     3.0e38f
#define EPS_W    1e-4f

static __device__ __forceinline__ unsigned umin2(unsigned a, unsigned b) { return a < b ? a : b; }
static __device__ __forceinline__ unsigned umax2(unsigned a, unsigned b) { return a > b ? a : b; }

// branch-free insert of key v into ascending 4-best {b0<=b1<=b2<=b3}
// (v_min_u32 / v_max_u32 compare-exchange chain, 7 VALU ops)
static __device__ __forceinline__ void ins_key(unsigned v,
                                               unsigned& b0, unsigned& b1,
                                               unsigned& b2, unsigned& b3) {
  b3 = umin2(b3, v);
  unsigned lo, hi;
  lo = umin2(b2, b3); hi = umax2(b2, b3); b2 = lo; b3 = hi;
  lo = umin2(b1, b2); hi = umax2(b1, b2); b1 = lo; b2 = hi;
  lo = umin2(b0, b1); hi = umax2(b0, b1); b0 = lo; b1 = hi;
}

__global__ __launch_bounds__(256)
void probe_blend_wmma(const float* __restrict__ xyz,
                      const float* __restrict__ probes,
                      const float* __restrict__ sh,
                      const int*   __restrict__ degp,
                      float*       __restrict__ out,
                      int n_points) {
  // --- stage SH tables in LDS (125 * 48 f32 = 24 KB) ---
  __shared__ float shs[NPROBE * SH_F];
  {
    float4* dst = (float4*)shs;
    const float4* src = (const float4*)sh;
    for (int i = threadIdx.x; i < NPROBE * SH_F / 4; i += blockDim.x) dst[i] = src[i];
  }
  __syncthreads();

  const int tid  = threadIdx.x;
  const int lane = tid & 31;
  const int wave = tid >> 5;
  const bool hiL = (lane >= 16);
  const int  col = lane & 15;
  const int  rowoff = hiL ? 8 : 0;      // C rows r map to probe r (lo) / r+8 (hi)
  const int  lim3 = (degp[0] + 1) * (degp[0] + 1) * 3;  // flat float limit for masking

  // --- A-matrix fragments: probes, K=4 rows (px,py,pz,|p|^2); pads = (0,0,0,BIG)
  // 32-bit A 16x4 layout: lanes 0-15 hold K0(v0),K1(v1); lanes 16-31 hold K2(v0),K3(v1)
  v2f aFrag[PTILES];
  #pragma unroll
  for (int t = 0; t < PTILES; ++t) {
    const int pi = t * 16 + col;
    float px = 0.f, py = 0.f, pz = 0.f, pp = BIGD;
    if (pi < NPROBE) {
      px = probes[pi * 3 + 0];
      py = probes[pi * 3 + 1];
      pz = probes[pi * 3 + 2];
      pp = px * px + py * py + pz * pz;
    }
    v2f a;
    if (!hiL) { a.x = px; a.y = py; }
    else      { a.x = pz; a.y = pp; }
    aFrag[t] = a;
  }

  const int waves_per_blk = blockDim.x >> 5;
  const int gwave   = blockIdx.x * waves_per_blk + wave;
  const int nwaves  = gridDim.x * waves_per_blk;
  const int ngroups = (n_points + 15) >> 4;

  for (int g = gwave; g < ngroups; g += nwaves) {
    const int base  = g << 4;
    const int pt    = base + col;
    const bool valid = (pt < n_points);

    float qx = 0.f, qy = 0.f, qz = 0.f;
    if (valid) {
      qx = xyz[pt * 3 + 0];
      qy = xyz[pt * 3 + 1];
      qz = xyz[pt * 3 + 2];
    }
    const float xx1 = qx * qx + qy * qy + qz * qz + 1.0f; // |x|^2 + 1 (keeps d^2 key > 0)

    // B-matrix 4x16: columns = points, rows (-2x,-2y,-2z,1)
    v2f b;
    if (!hiL) { b.x = -2.f * qx; b.y = -2.f * qy; }
    else      { b.x = -2.f * qz; b.y = 1.f; }

    // --- WMMA: D = |p|^2 - 2 x.p + (|x|^2 + 1)  ==  d^2 + 1  (positive) ---
    // top-4 kept as sortable u32 keys: (float bits & ~0x7F) | probe_idx
    unsigned b0 = 0xFFFFFFFFu, b1 = 0xFFFFFFFFu, b2 = 0xFFFFFFFFu, b3 = 0xFFFFFFFFu;
    #pragma unroll
    for (int t = 0; t < PTILES; ++t) {
      v8f c = { xx1, xx1, xx1, xx1, xx1, xx1, xx1, xx1 };
      c = __builtin_amdgcn_wmma_f32_16x16x4_f32(
              /*neg_a=*/false, aFrag[t], /*neg_b=*/false, b,
              /*c_mod=*/(short)0, c, /*reuse_a=*/false, /*reuse_b=*/false);
      #pragma unroll
      for (int r = 0; r < 8; ++r) {
        const unsigned key = (__float_as_uint(c[r]) & 0xFFFFFF80u) |
                             (unsigned)(t * 16 + r + rowoff);
        ins_key(key, b0, b1, b2, b3);
      }
    }

    // --- merge the lane pair (each held 64 of the 128 probe slots) ---
    ins_key((unsigned)__shfl_xor((int)b0, 16, 32), b0, b1, b2, b3);
    ins_key((unsigned)__shfl_xor((int)b1, 16, 32), b0, b1, b2, b3);
    ins_key((unsigned)__shfl_xor((int)b2, 16, 32), b0, b1, b2, b3);
    ins_key((unsigned)__shfl_xor((int)b3, 16, 32), b0, b1, b2, b3);

    // --- unpack; d = sqrt(max(key_float - 1, 0)); w = 1/(d+eps), normalized ---
    const int   p0 = (int)(b0 & 127u), p1 = (int)(b1 & 127u),
                p2 = (int)(b2 & 127u), p3 = (int)(b3 & 127u);
    const float s0 = sqrtf(fmaxf(__uint_as_float(b0 & 0xFFFFFF80u) - 1.0f, 0.f));
    const float s1 = sqrtf(fmaxf(__uint_as_float(b1 & 0xFFFFFF80u) - 1.0f, 0.f));
    const float s2 = sqrtf(fmaxf(__uint_as_float(b2 & 0xFFFFFF80u) - 1.0f, 0.f));
    const float s3 = sqrtf(fmaxf(__uint_as_float(b3 & 0xFFFFFF80u) - 1.0f, 0.f));
    float w0 = 1.f / (s0 + EPS_W);
    float w1 = 1.f / (s1 + EPS_W);
    float w2 = 1.f / (s2 + EPS_W);
    float w3 = 1.f / (s3 + EPS_W);
    const float inv = 1.f / (w0 + w1 + w2 + w3);
    w0 *= inv; w1 *= inv; w2 *= inv; w3 *= inv;

    // --- blend half the SH row per lane (lo lane: floats 0..23, hi: 24..47) ---
    if (valid) {
      const int half = hiL ? 24 : 0;
      float4 acc[6];
      #pragma unroll
      for (int j = 0; j < 6; ++j) acc[j] = make_float4(0.f, 0.f, 0.f, 0.f);

      const int   pidx[4] = { p0, p1, p2, p3 };
      const float wv[4]   = { w0, w1, w2, w3 };
      #pragma unroll
      for (int q = 0; q < 4; ++q) {
        const float4* row = (const float4*)&shs[pidx[q] * SH_F + half];
        const float w = wv[q];
        #pragma unroll
        for (int j = 0; j < 6; ++j) {
          const float4 v = row[j];   // ds_load_b128
          acc[j].x += w * v.x; acc[j].y += w * v.y;
          acc[j].z += w * v.z; acc[j].w += w * v.w;
        }
      }

      float* op = out + (size_t)pt * SH_F + half;
      #pragma unroll
      for (int j = 0; j < 6; ++j) {
        const int f0 = half + j * 4;  // flat float index of .x
        float4 v = acc[j];
        v.x = (f0 + 0 < lim3) ? v.x : 0.f;
        v.y = (f0 + 1 < lim3) ? v.y : 0.f;
        v.z = (f0 + 2 < lim3) ? v.z : 0.f;
        v.w = (f0 + 3 < lim3) ? v.w : 0.f;
        ((float4*)op)[j] = v;        // global_store_b128
      }
    }
  }
}

extern "C" void kernel_launch(void* const* d_in, const int* in_sizes, int n_in,
                              void* d_out, int out_size, void* d_ws, size_t ws_size,
                              hipStream_t stream) {
  const float* xyz    = (const float*)d_in[0];   // [N,3] f32
  const float* probes = (const float*)d_in[1];   // [125,3] f32
  const float* sh     = (const float*)d_in[2];   // [125,16,3] f32
  const int*   deg    = (const int*)  d_in[3];   // scalar
  float*       out    = (float*)d_out;           // [N,16,3] f32

  const int n_points = in_sizes[0] / 3;
  const int threads  = 256;                       // 8 wave32s
  const int blocks   = 1024;                      // grid-stride over 125k groups

  probe_blend_wmma<<<blocks, threads, 0, stream>>>(xyz, probes, sh, deg, out, n_points);
}